// Group_17738214933230
// MI455X (gfx1250) — compile-verified
//
#include <hip/hip_runtime.h>

#define B_   32
#define N_   8192
#define G_   512
#define K_   32
#define GT   16      // centers per block tile
#define CH   512     // points per chunk
#define BIGF 1e10f
#define INFF 3.0e38f

typedef __attribute__((ext_vector_type(2))) float v2f;
typedef __attribute__((ext_vector_type(8))) float v8f;

// ---------------------------------------------------------------------------
// Kernel 1: farthest point sampling. One block per batch; points + min-dist
// live in registers (8 per thread); 512 rounds of 2-level wave32 argmax.
// Writes center coordinates [G,3] for its batch.
// ---------------------------------------------------------------------------
__global__ __launch_bounds__(1024) void fps_kernel(const float* __restrict__ xyz,
                                                   float* __restrict__ outc)
{
    const int b = blockIdx.x;
    const float* xb = xyz + (size_t)b * N_ * 3;
    float* cb = outc + (size_t)b * G_ * 3;

    const int t = threadIdx.x;
    const int lane = t & 31, wv = t >> 5;

    float px[8], py[8], pz[8], md[8];
#pragma unroll
    for (int j = 0; j < 8; ++j) {
        int n = t + j * 1024;
        px[j] = xb[3 * n + 0];
        py[j] = xb[3 * n + 1];
        pz[j] = xb[3 * n + 2];
        md[j] = BIGF;
    }

    __shared__ float rD[32];
    __shared__ int   rI[32];
    __shared__ float cvec[3];
    __shared__ int   sFar;
    if (t == 0) sFar = 0;
    __syncthreads();

    for (int it = 0; it < G_; ++it) {
        if (t == 0) {
            int f = sFar;                       // farthest BEFORE update (idxs[it])
            float cx = xb[3 * f + 0], cy = xb[3 * f + 1], cz = xb[3 * f + 2];
            cvec[0] = cx; cvec[1] = cy; cvec[2] = cz;
            cb[3 * it + 0] = cx; cb[3 * it + 1] = cy; cb[3 * it + 2] = cz;
        }
        __syncthreads();
        float cx = cvec[0], cy = cvec[1], cz = cvec[2];

        float bd = -1.0f; int bi = 0x7fffffff;
#pragma unroll
        for (int j = 0; j < 8; ++j) {
            float dx = px[j] - cx, dy = py[j] - cy, dz = pz[j] - cz;
            float d = dx * dx + dy * dy + dz * dz;
            md[j] = fminf(md[j], d);
            int n = t + j * 1024;
            if (md[j] > bd || (md[j] == bd && n < bi)) { bd = md[j]; bi = n; }
        }
        // wave32 argmax (ties -> smaller index, matching jnp.argmax)
#pragma unroll
        for (int off = 16; off > 0; off >>= 1) {
            float od = __shfl_xor(bd, off);
            int   oi = __shfl_xor(bi, off);
            if (od > bd || (od == bd && oi < bi)) { bd = od; bi = oi; }
        }
        if (lane == 0) { rD[wv] = bd; rI[wv] = bi; }
        __syncthreads();
        if (wv == 0) {
            float d2 = rD[lane]; int i2 = rI[lane];
#pragma unroll
            for (int off = 16; off > 0; off >>= 1) {
                float od = __shfl_xor(d2, off);
                int   oi = __shfl_xor(i2, off);
                if (od > d2 || (od == d2 && oi < i2)) { d2 = od; i2 = oi; }
            }
            if (lane == 0) sFar = i2;
        }
        __syncthreads();
    }
}

// ---------------------------------------------------------------------------
// Kernel 2: WMMA distance tiles + streaming wave-parallel top-32 + gather.
// Block = (batch, tile of 16 centers), 512 threads = 16 waves.
// dist[m][n] = ||c_m||^2 + ||x_n||^2 + WMMA((-2*c) x) ; norms folded at read.
// Wave w keeps center w's sorted top-32 in registers: lane i = rank i.
// ---------------------------------------------------------------------------
__global__ __launch_bounds__(512) void knn_kernel(const float* __restrict__ xyz,
                                                  const float* __restrict__ centers,
                                                  float* __restrict__ outn)
{
    const int b  = blockIdx.y;
    const int gt = blockIdx.x;                       // 0..G_/GT-1
    const float* xb = xyz + (size_t)b * N_ * 3;
    const float* cbase = centers + ((size_t)b * G_ + (size_t)gt * GT) * 3;

    const int t    = threadIdx.x;
    const int lane = t & 31, wv = t >> 5;            // 16 waves
    const int n16  = lane & 15;
    const int half = lane >> 4;                      // 0: K=0,1  1: K=2,3

    __shared__ float tile[GT * CH];                  // 32 KB: -2*c.x subtile results
    __shared__ float xn[CH];                         // point squared norms
    __shared__ float scx[GT], scy[GT], scz[GT], scn[GT];

    if (t < GT) {
        float cx = cbase[3 * t + 0], cy = cbase[3 * t + 1], cz = cbase[3 * t + 2];
        scx[t] = cx; scy[t] = cy; scz[t] = cz;
        scn[t] = cx * cx + cy * cy + cz * cz;
    }
    __syncthreads();

    // A matrix 16x4 (M x K): lane holds row M = lane%16; halves hold K={0,1}/{2,3}.
    // K-dim = (x, y, z, 0), scaled by -2.
    v2f A;
    A[0] = -2.0f * (half ? scz[n16] : scx[n16]);
    A[1] = half ? 0.0f : (-2.0f * scy[n16]);

    const float cn = scn[wv];                        // this wave's center norm

    float listD = INFF;                              // lane i: rank-i distance
    int   listI = 0x7fffffff;                        // lane i: rank-i point index

    for (int ck = 0; ck < N_ / CH; ++ck) {
        const int base = ck * CH;

        // point norms for this chunk
        {
            int p = base + t;
            float x = xb[3 * p + 0], y = xb[3 * p + 1], z = xb[3 * p + 2];
            xn[t] = x * x + y * y + z * z;
        }

        // each wave computes two 16x16 subtiles via V_WMMA_F32_16X16X4_F32
#pragma unroll
        for (int s2 = 0; s2 < 2; ++s2) {
            int s   = wv * 2 + s2;                   // subtile 0..31
            int col = s * 16 + n16;                  // chunk column 0..511
            const float* pp = xb + 3 * (size_t)(base + col);
            // B matrix 4x16 (K x N): VGPR r holds rows K=r (lanes 0-15) / K=r+2
            v2f Bv;
            Bv[0] = half ? pp[2] : pp[0];            // K=2 (z) | K=0 (x)
            Bv[1] = half ? 0.0f  : pp[1];            // K=3 (0) | K=1 (y)
            v8f Cc = {};
            v8f D = __builtin_amdgcn_wmma_f32_16x16x4_f32(
                        false, A, false, Bv, (short)0, Cc, false, false);
#pragma unroll
            for (int r = 0; r < 8; ++r) {
                int m = r + (half ? 8 : 0);          // C/D: VGPR r -> M=r / M=r+8
                tile[m * CH + col] = D[r];
            }
        }
        __syncthreads();

        // streaming top-32 merge: wave wv handles center row wv
        unsigned consumed = 0;
        float lmin; int lidx; int lj;
        auto rescan = [&]() {
            lmin = INFF; lidx = 0x7fffffff; lj = -1;
#pragma unroll
            for (int j = 0; j < CH / 32; ++j) {
                if (consumed & (1u << j)) continue;
                int c = j * 32 + lane;
                float d = tile[wv * CH + c] + xn[c] + cn;
                int pi = base + c;
                if (d < lmin || (d == lmin && pi < lidx)) { lmin = d; lidx = pi; lj = j; }
            }
        };
        rescan();

        for (int e = 0; e < K_; ++e) {
            // global (dist,idx) min across the wave
            float gd = lmin; int gi = lidx;
#pragma unroll
            for (int off = 16; off > 0; off >>= 1) {
                float od = __shfl_xor(gd, off);
                int   oi = __shfl_xor(gi, off);
                if (od < gd || (od == gd && oi < gi)) { gd = od; gi = oi; }
            }
            float kD = __shfl(listD, 31);
            int   kI = __shfl(listI, 31);
            if (!(gd < kD || (gd == kD && gi < kI))) break;   // nothing else can enter

            // O(1) sorted insert: position = popcount(lanes lexicographically less)
            bool lessv = (listD < gd) || (listD == gd && listI < gi);
            int  p = __popcll(__ballot(lessv));
            float uD = __shfl_up(listD, 1);
            int   uI = __shfl_up(listI, 1);
            if (lane == p)      { listD = gd; listI = gi; }
            else if (lane > p)  { listD = uD; listI = uI; }

            // owning lane consumes the extracted candidate and rescans
            if (lmin == gd && lidx == gi) { consumed |= (1u << lj); rescan(); }
        }
        __syncthreads();                             // tile/xn reused next chunk
    }

    // lane i of wave w writes neighbor rank i of center gt*16+w (gather - center)
    {
        int g = gt * GT + wv;
        const float* pp = xb + 3 * (size_t)listI;
        float ox = pp[0] - scx[wv];
        float oy = pp[1] - scy[wv];
        float oz = pp[2] - scz[wv];
        size_t o = (((size_t)b * G_ + g) * K_ + lane) * 3;
        outn[o + 0] = ox; outn[o + 1] = oy; outn[o + 2] = oz;
    }
}

// ---------------------------------------------------------------------------
extern "C" void kernel_launch(void* const* d_in, const int* in_sizes, int n_in,
                              void* d_out, int out_size, void* d_ws, size_t ws_size,
                              hipStream_t stream)
{
    const float* xyz = (const float*)d_in[0];
    float* out        = (float*)d_out;
    float* out_center = out + (size_t)B_ * G_ * K_ * 3;   // tuple: [neigh | center]

    fps_kernel<<<B_, 1024, 0, stream>>>(xyz, out_center);

    dim3 grid(G_ / GT, B_);
    knn_kernel<<<grid, 512, 0, stream>>>(xyz, out_center, out);
}